// FOEDDPLL_23106924052734
// MI455X (gfx1250) — compile-verified
//
#include <hip/hip_runtime.h>
#include <math.h>

// ---------------------------------------------------------------------------
// FOE/DD-PLL carrier phase recovery (serial nonlinear recurrence).
// 1 block x 32 threads (one wave32). Lanes 0..1 run the two independent mode
// chains; all 32 lanes cooperatively stage input tiles into LDS using CDNA5
// async global->LDS copies (ASYNCcnt), double-buffered so global latency is
// fully overlapped with the ALU-latency-bound serial loop.
// ---------------------------------------------------------------------------

#define TILE 512
#define NM 2

#if defined(__has_builtin)
#  if __has_builtin(__builtin_amdgcn_global_load_async_to_lds_b128)
#    define ASYNC_COPY 1
#  endif
#endif

typedef int v4i __attribute__((vector_size(16)));
typedef __attribute__((address_space(1))) v4i as1_v4i;   // global (__device__) v4i
typedef __attribute__((address_space(3))) v4i as3_v4i;   // LDS (__shared__) v4i

__launch_bounds__(32, 1)
__global__ void foeddpll_kernel(const float* __restrict__ Ei_re,
                                const float* __restrict__ Ei_im,
                                const float* __restrict__ Tx_re,
                                const float* __restrict__ Tx_im,
                                const float* __restrict__ eta_n,
                                const float* __restrict__ eta_f,
                                const int*   __restrict__ pmask,
                                float* __restrict__ out,
                                int nsymb)
{
    __shared__ __align__(16) float sh_yre[2 * TILE * NM];
    __shared__ __align__(16) float sh_yim[2 * TILE * NM];
    __shared__ __align__(16) float sh_xre[2 * TILE * NM];
    __shared__ __align__(16) float sh_xim[2 * TILE * NM];
    __shared__ __align__(16) int   sh_msk[2 * TILE * NM];

    const int lane = (int)threadIdx.x;

    const float Kn = tanhf(eta_n[0]);
    const float Kf = tanhf(eta_f[0]);

    const float S10    = 3.16227766016837933f;   // sqrt(10)
    const float IS10   = 0.316227766016837933f;  // 1/sqrt(10)
    const float P2     = 1.57079632679489662f;   // pi/2  (unwrap period)
    const float P4     = 0.785398163397448310f;  // pi/4  (discont)
    const float INV_P2 = 0.636619772367581343f;  // 2/pi

    const int ntiles = nsymb / TILE;
    const int rem    = nsymb - ntiles * TILE;

    // ---- stage one tile of all 5 input streams into LDS buffer `buf` ----
    auto stage = [&](int buf, int t) {
        const size_t foff = (size_t)t   * (TILE * NM);   // elements
        const size_t boff = (size_t)buf * (TILE * NM);
#if ASYNC_COPY
        const char* g0 = (const char*)(Ei_re + foff) + lane * 16;
        const char* g1 = (const char*)(Ei_im + foff) + lane * 16;
        const char* g2 = (const char*)(Tx_re + foff) + lane * 16;
        const char* g3 = (const char*)(Tx_im + foff) + lane * 16;
        const char* g4 = (const char*)(pmask + foff) + lane * 16;
        char* l0 = (char*)(sh_yre + boff) + lane * 16;
        char* l1 = (char*)(sh_yim + boff) + lane * 16;
        char* l2 = (char*)(sh_xre + boff) + lane * 16;
        char* l3 = (char*)(sh_xim + boff) + lane * 16;
        char* l4 = (char*)(sh_msk + boff) + lane * 16;
        // 8 iterations x 5 streams = 40 async B128 copies (32 lanes x 16B each)
        #pragma unroll
        for (int i = 0; i < (TILE * NM * 4) / 512; ++i) {
            __builtin_amdgcn_global_load_async_to_lds_b128((as1_v4i*)(g0 + i * 512), (as3_v4i*)(l0 + i * 512), 0, 0);
            __builtin_amdgcn_global_load_async_to_lds_b128((as1_v4i*)(g1 + i * 512), (as3_v4i*)(l1 + i * 512), 0, 0);
            __builtin_amdgcn_global_load_async_to_lds_b128((as1_v4i*)(g2 + i * 512), (as3_v4i*)(l2 + i * 512), 0, 0);
            __builtin_amdgcn_global_load_async_to_lds_b128((as1_v4i*)(g3 + i * 512), (as3_v4i*)(l3 + i * 512), 0, 0);
            __builtin_amdgcn_global_load_async_to_lds_b128((as1_v4i*)(g4 + i * 512), (as3_v4i*)(l4 + i * 512), 0, 0);
        }
#else
        const float4* G0 = (const float4*)(Ei_re + foff);
        const float4* G1 = (const float4*)(Ei_im + foff);
        const float4* G2 = (const float4*)(Tx_re + foff);
        const float4* G3 = (const float4*)(Tx_im + foff);
        const float4* G4 = (const float4*)(pmask + foff);
        float4* L0 = (float4*)(sh_yre + boff);
        float4* L1 = (float4*)(sh_yim + boff);
        float4* L2 = (float4*)(sh_xre + boff);
        float4* L3 = (float4*)(sh_xim + boff);
        float4* L4 = (float4*)(sh_msk + boff);
        for (int i = lane; i < (TILE * NM) / 4; i += 32) {
            L0[i] = G0[i]; L1[i] = G1[i]; L2[i] = G2[i]; L3[i] = G3[i]; L4[i] = G4[i];
        }
        __syncthreads();
#endif
        // prefetch the tile after the one just staged (speculative, L2-bound)
        if ((t + 1) * TILE + TILE <= nsymb) {
            const size_t poff = foff + (size_t)(TILE * NM);
            __builtin_prefetch((const char*)(Ei_re + poff) + lane * 128, 0, 3);
            __builtin_prefetch((const char*)(Ei_im + poff) + lane * 128, 0, 3);
            __builtin_prefetch((const char*)(Tx_re + poff) + lane * 128, 0, 3);
            __builtin_prefetch((const char*)(Tx_im + poff) + lane * 128, 0, 3);
            __builtin_prefetch((const char*)(pmask + poff) + lane * 128, 0, 3);
        }
    };

    // ---- per-mode PLL carry state (meaningful in lanes 0..1) ----
    float u0 = 0.f, u2 = 0.f, w = 0.f, phi = 0.f, prev = 0.f;
    int q = 0;  // unwrap correction in quadrants (multiples of pi/2)

    auto step = [&](float yre, float yim, float xre, float xim, int mk, int k, int m) {
        float phip = phi + w;
        float sn, cs;
        __sincosf(phip, &sn, &cs);
        // r = y * exp(-i*phip)
        float rre = fmaf(yre, cs,  yim * sn);
        float rim = fmaf(yim, cs, -(yre * sn));
        // separable 16-QAM slicer: nearest of {-3,-1,1,3}/sqrt(10) per axis
        float qre = fmaf(fminf(fmaxf(rintf(fmaf(rre, S10, 3.f) * 0.5f), 0.f), 3.f), 2.f, -3.f) * IS10;
        float qim = fmaf(fminf(fmaxf(rintf(fmaf(rim, S10, 3.f) * 0.5f), 0.f), 3.f), 2.f, -3.f) * IS10;
        float tre = mk ? qre : xre;   // mask==1: decision-directed, else pilot
        float tim = mk ? qim : xim;
        // d/dphi |y e^{-i phi} - t|^2 = -2 Im(r * conj(t))
        float gk = -2.f * (rim * tre - rre * tim);
        u0 = u0 - u2 + gk;            // u1 is dead state: u0 <- u0 - old_u2 + gk
        u2 = gk;
        phi = fmaf(-Kn, u0, phip);
        w   = fmaf(-Kf, u0, w);
        // unwrap(theta, period=pi/2): corrections are exact multiples of pi/2
        float dd = phip - prev;
        prev = phip;
        if (fabsf(dd) >= P4) {
            float ddm = fmodf(dd + P4, P2);
            if (ddm < 0.f) ddm += P2;
            ddm -= P4;
            if (ddm == -P4 && dd > 0.f) ddm = P4;
            q += (int)rintf((ddm - dd) * INV_P2);
        }
        // exp(i*(phip + q*pi/2)) = quadrant-rotated (cs, sn): no 2nd sincos
        int qa = q & 3;
        float c2 = (qa == 0) ? cs : (qa == 1) ? -sn : (qa == 2) ? -cs :  sn;
        float s2 = (qa == 0) ? sn : (qa == 1) ?  cs : (qa == 2) ? -sn : -cs;
        float ore = fmaf(yre, c2, -(yim * s2));
        float oim = fmaf(yre, s2,   yim * c2);
        size_t o = (size_t)4 * (size_t)k + 2 * (size_t)m;
        out[o]     = ore;
        out[o + 1] = oim;
    };

    // ---- main tiled loop: double-buffered async staging ----
    if (ntiles > 0) stage(0, 0);
    int cur = 0;
    for (int t = 0; t < ntiles; ++t) {
        if (t + 1 < ntiles) {
#if ASYNC_COPY
            asm volatile("s_wait_dscnt 0x0" ::: "memory");   // old reads drained
#endif
            stage(cur ^ 1, t + 1);                           // prefetch next tile
#if ASYNC_COPY
            asm volatile("s_wait_asynccnt 0x28" ::: "memory"); // <=40 pending => tile t done
#endif
        } else {
#if ASYNC_COPY
            asm volatile("s_wait_asynccnt 0x0" ::: "memory");
#endif
        }
        if (lane < NM) {
            const int m = lane;
            const int b = cur * (TILE * NM);
            #pragma unroll 4
            for (int j = 0; j < TILE; ++j) {
                int s = b + 2 * j + m;
                step(sh_yre[s], sh_yim[s], sh_xre[s], sh_xim[s], sh_msk[s],
                     t * TILE + j, m);
            }
        }
        cur ^= 1;
    }

    // ---- tail (nsymb % TILE symbols) straight from global (L2-hot) ----
    if (rem > 0 && lane < NM) {
        for (int j = 0; j < rem; ++j) {
            int k = ntiles * TILE + j;
            int s = 2 * k + lane;
            step(Ei_re[s], Ei_im[s], Tx_re[s], Tx_im[s], pmask[s], k, lane);
        }
    }
}

extern "C" void kernel_launch(void* const* d_in, const int* in_sizes, int n_in,
                              void* d_out, int out_size, void* d_ws, size_t ws_size,
                              hipStream_t stream) {
    (void)n_in; (void)out_size; (void)d_ws; (void)ws_size;
    const float* Ei_re = (const float*)d_in[0];
    const float* Ei_im = (const float*)d_in[1];
    const float* Tx_re = (const float*)d_in[2];
    const float* Tx_im = (const float*)d_in[3];
    const float* eta_n = (const float*)d_in[4];
    const float* eta_f = (const float*)d_in[5];
    const int*   pmask = (const int*)d_in[6];
    float* out = (float*)d_out;
    const int nsymb = in_sizes[0] / 2;   // [Nsymb, 2] flattened

    foeddpll_kernel<<<1, 32, 0, stream>>>(Ei_re, Ei_im, Tx_re, Tx_im,
                                          eta_n, eta_f, pmask, out, nsymb);
}